// GIN_61607010894468
// MI455X (gfx1250) — compile-verified
//
#include <hip/hip_runtime.h>

#define N_NODES 100000
#define N_EDGES 1600000
#define D_FEAT  128
#define D_HID   512
#define D_OUT   128
#define SCAN_B  1024
#define NB      ((N_NODES + SCAN_B - 1) / SCAN_B)   // 98 scan blocks

typedef float v2f __attribute__((ext_vector_type(2)));
typedef float v8f __attribute__((ext_vector_type(8)));

__device__ __forceinline__ v8f wmma4(v2f a, v2f b, v8f c) {
  return __builtin_amdgcn_wmma_f32_16x16x4_f32(false, a, false, b,
                                               (short)0, c, false, false);
}

// ---------------- CSR construction ----------------

__global__ void k_zero(int* __restrict__ p, int n) {
  int i = blockIdx.x * blockDim.x + threadIdx.x;
  if (i < n) p[i] = 0;
}

__global__ void k_deg(const int* __restrict__ dst, int* __restrict__ deg) {
  int e = blockIdx.x * blockDim.x + threadIdx.x;
  if (e < N_EDGES) atomicAdd(&deg[dst[e]], 1);
}

// block-level inclusive scan (Hillis-Steele), emits block totals
__global__ __launch_bounds__(SCAN_B)
void k_scan1(const int* __restrict__ deg, int* __restrict__ incl,
             int* __restrict__ bsum, int n) {
  __shared__ int s[SCAN_B];
  int i = blockIdx.x * SCAN_B + threadIdx.x;
  s[threadIdx.x] = (i < n) ? deg[i] : 0;
  __syncthreads();
  for (int off = 1; off < SCAN_B; off <<= 1) {
    int t = (threadIdx.x >= off) ? s[threadIdx.x - off] : 0;
    __syncthreads();
    s[threadIdx.x] += t;
    __syncthreads();
  }
  if (i < n) incl[i] = s[threadIdx.x];
  if (threadIdx.x == SCAN_B - 1) bsum[blockIdx.x] = s[SCAN_B - 1];
}

// serial exclusive scan over <=128 block totals (negligible work)
__global__ void k_scan2(int* __restrict__ bsum, int nb) {
  if (threadIdx.x == 0 && blockIdx.x == 0) {
    int run = 0;
    for (int i = 0; i < nb; ++i) { int t = bsum[i]; bsum[i] = run; run += t; }
  }
}

// offs[i+1] = global inclusive scan; offs[0] = 0
__global__ __launch_bounds__(SCAN_B)
void k_scan3(const int* __restrict__ incl, const int* __restrict__ bsum,
             int* __restrict__ offs, int n) {
  int i = blockIdx.x * SCAN_B + threadIdx.x;
  if (i < n) offs[i + 1] = incl[i] + bsum[blockIdx.x];
  if (i == 0) offs[0] = 0;
}

__global__ void k_bin(const int* __restrict__ src, const int* __restrict__ dst,
                      const int* __restrict__ offs, int* __restrict__ cursor,
                      int* __restrict__ bsrc) {
  int e = blockIdx.x * blockDim.x + threadIdx.x;
  if (e >= N_EDGES) return;
  int d = dst[e];
  int pos = atomicAdd(&cursor[d], 1);
  bsrc[offs[d] + pos] = src[e];
}

// ---------------- Aggregation: h[v] = x[v] + sum_{e in bin(v)} x[src(e)] ----------------
// one wave32 per node; lane l owns features [4l, 4l+4). Pure gathers, zero float atomics.
__global__ __launch_bounds__(256)
void k_aggr(const float* __restrict__ x, const int* __restrict__ offs,
            const int* __restrict__ bsrc, float* __restrict__ h) {
  int node = (blockIdx.x << 3) + (threadIdx.x >> 5);
  if (node >= N_NODES) return;
  int lane = threadIdx.x & 31;
  float4 acc = *reinterpret_cast<const float4*>(x + (size_t)node * D_FEAT + lane * 4);
  int e0 = offs[node], e1 = offs[node + 1];
  for (int e = e0; e < e1; ++e) {
    int s = bsrc[e];
    float4 v = *reinterpret_cast<const float4*>(x + (size_t)s * D_FEAT + lane * 4);
    acc.x += v.x; acc.y += v.y; acc.z += v.z; acc.w += v.w;
  }
  *reinterpret_cast<float4*>(h + (size_t)node * D_FEAT + lane * 4) = acc;
}

// ---------------- Fused MLP: out = relu(h@W1+b1)@W2+b2 ----------------
// 32 nodes/block (two 16-row M-tiles), 256 threads = 8 waves, f32 WMMA 16x16x4.
// B fragments are loaded once per k-step and reused across both M-tiles.
__global__ __launch_bounds__(256)
void gin_mlp(const float* __restrict__ h,
             const float* __restrict__ W1, const float* __restrict__ b1,
             const float* __restrict__ W2, const float* __restrict__ b2,
             float* __restrict__ out) {
  extern __shared__ float smem[];
  float* sH  = smem;                 // 32 x 128 = 16 KB
  float* sH1 = smem + 32 * D_FEAT;   // 32 x 512 = 64 KB

  const int tid   = threadIdx.x;
  const int wave  = tid >> 5;          // 0..7
  const int lane  = tid & 31;
  const int row16 = lane & 15;
  const int khalf = (lane >> 4) << 1;  // K sub-offset: 0 (lo lanes) / 2 (hi lanes)
  const int rowhi = (lane >> 4) << 3;  // C/D: VGPR i holds M=i+rowhi
  const size_t base = (size_t)blockIdx.x * 32 * D_FEAT;

  for (int i = tid; i < 32 * D_FEAT; i += 256) sH[i] = h[base + i];
  __syncthreads();

  // ---- GEMM1 (32x128)@(128x512), bias+relu -> sH1 ----
  for (int t = 0; t < 4; ++t) {
    const int n0  = (wave + (t << 3)) << 4;
    const int col = n0 + row16;
    v8f acc0 = {0.f,0.f,0.f,0.f,0.f,0.f,0.f,0.f};
    v8f acc1 = {0.f,0.f,0.f,0.f,0.f,0.f,0.f,0.f};
#pragma unroll
    for (int kk = 0; kk < D_FEAT; kk += 4) {
      const int kb = kk + khalf;
      v2f b;
      b.x = W1[(size_t)kb * D_HID + col];
      b.y = W1[(size_t)(kb + 1) * D_HID + col];
      v2f a0 = *reinterpret_cast<const v2f*>(sH + row16 * D_FEAT + kk + khalf);
      v2f a1 = *reinterpret_cast<const v2f*>(sH + (16 + row16) * D_FEAT + kk + khalf);
      acc0 = wmma4(a0, b, acc0);
      acc1 = wmma4(a1, b, acc1);
    }
    const float bias = b1[col];
#pragma unroll
    for (int i = 0; i < 8; ++i) {
      float v0 = acc0[i] + bias;
      float v1 = acc1[i] + bias;
      sH1[(i + rowhi) * D_HID + col]      = v0 > 0.f ? v0 : 0.f;
      sH1[(16 + i + rowhi) * D_HID + col] = v1 > 0.f ? v1 : 0.f;
    }
  }
  __syncthreads();

  // ---- GEMM2 (32x512)@(512x128) + b2 -> out ----
  {
    const int n0  = wave << 4;
    const int col = n0 + row16;
    v8f acc0 = {0.f,0.f,0.f,0.f,0.f,0.f,0.f,0.f};
    v8f acc1 = {0.f,0.f,0.f,0.f,0.f,0.f,0.f,0.f};
#pragma unroll 8
    for (int kk = 0; kk < D_HID; kk += 4) {
      const int kb = kk + khalf;
      v2f b;
      b.x = W2[(size_t)kb * D_OUT + col];
      b.y = W2[(size_t)(kb + 1) * D_OUT + col];
      v2f a0 = *reinterpret_cast<const v2f*>(sH1 + row16 * D_HID + kk + khalf);
      v2f a1 = *reinterpret_cast<const v2f*>(sH1 + (16 + row16) * D_HID + kk + khalf);
      acc0 = wmma4(a0, b, acc0);
      acc1 = wmma4(a1, b, acc1);
    }
    const float bias = b2[col];
    const size_t obase = (size_t)blockIdx.x * 32 * D_OUT;
#pragma unroll
    for (int i = 0; i < 8; ++i) {
      out[obase + (size_t)(i + rowhi) * D_OUT + col]      = acc0[i] + bias;
      out[obase + (size_t)(16 + i + rowhi) * D_OUT + col] = acc1[i] + bias;
    }
  }
}

extern "C" void kernel_launch(void* const* d_in, const int* in_sizes, int n_in,
                              void* d_out, int out_size, void* d_ws, size_t ws_size,
                              hipStream_t stream) {
  const float* x  = (const float*)d_in[0];
  const int*   ei = (const int*)d_in[1];   // edge_index [2, N_EDGES]
  const float* W1 = (const float*)d_in[2];
  const float* b1 = (const float*)d_in[3];
  const float* W2 = (const float*)d_in[4];
  const float* b2 = (const float*)d_in[5];
  float* out = (float*)d_out;

  const int* src = ei;
  const int* dst = ei + N_EDGES;

  // workspace layout (all 4B-aligned; h is 16B-aligned at base)
  float* h    = (float*)d_ws;                       // N_NODES*128 f32
  int*  bsrc  = (int*)(h + (size_t)N_NODES * D_FEAT); // N_EDGES
  int*  deg   = bsrc + N_EDGES;                     // N_NODES
  int*  curs  = deg + N_NODES;                      // N_NODES
  int*  incl  = curs + N_NODES;                     // N_NODES
  int*  offs  = incl + N_NODES;                     // N_NODES+1
  int*  bsum  = offs + N_NODES + 1;                 // NB

  // 1) CSR build: deg -> scan -> bin  (3.2M int atomics total)
  k_zero<<<(2 * N_NODES + 255) / 256, 256, 0, stream>>>(deg, 2 * N_NODES); // deg+curs contiguous
  k_deg<<<(N_EDGES + 255) / 256, 256, 0, stream>>>(dst, deg);
  k_scan1<<<NB, SCAN_B, 0, stream>>>(deg, incl, bsum, N_NODES);
  k_scan2<<<1, 32, 0, stream>>>(bsum, NB);
  k_scan3<<<NB, SCAN_B, 0, stream>>>(incl, bsum, offs, N_NODES);
  k_bin<<<(N_EDGES + 255) / 256, 256, 0, stream>>>(src, dst, offs, curs, bsrc);

  // 2) aggregation: pure L2-resident gathers, no float atomics
  k_aggr<<<(N_NODES + 7) / 8, 256, 0, stream>>>(x, offs, bsrc, h);

  // 3) fused MLP with f32 WMMA, 80 KB dynamic LDS
  gin_mlp<<<N_NODES / 32, 256, 80 * 1024, stream>>>(h, W1, b1, W2, b2, out);
}